// GHMCLoss_76922864271904
// MI455X (gfx1250) — compile-verified
//
#include <hip/hip_runtime.h>

// GHM-C loss, fused single-pass formulation:
//   loss = (1/max(n,1)) * sum_b S_b / c_b      (tot cancels exactly)
//
// Pass 1: stream input once via gfx1250 GLOBAL_LOAD_ASYNC_TO_LDS_B128 with a
//         4-deep ring (issue-ahead 3, s_wait_asynccnt 3) so each wave keeps
//         3 x 512B async loads in flight; fallback is direct NT b128 loads +
//         global_prefetch_b8. Bin into per-thread private LDS
//         float2{count,sum} slots (conflict-free padded layout, no atomics ->
//         deterministic), tree-reduce per block, write partials to d_ws.
// Pass 2: one 10-wave block reduces 10 bins across blocks (wave32 shuffles).

#define BINS 10
#define C_CLS 80
#define T1 256        // threads per block, pass 1 (8 waves of 32)
#define PAD 11        // float2 slots per thread: 88B stride -> 22-dword lane
                      // stride, conflict-free across 32 lanes
#define DEPTH 4       // async staging ring depth (issue-ahead = 3)
#define NB_MAX 1024   // pass-1 blocks (partials: 1024*20 floats = 80 KB in ws)

typedef float v4f __attribute__((ext_vector_type(4)));
typedef int   vi4 __attribute__((vector_size(16)));   // matches builtin params
typedef __attribute__((address_space(1))) vi4* as1_vi4_ptr;   // global int4*
typedef __attribute__((address_space(3))) vi4* as3_vi4_ptr;   // LDS int4*

#if defined(__HIP_DEVICE_COMPILE__) && \
    __has_builtin(__builtin_amdgcn_global_load_async_to_lds_b128)
#define HAVE_ASYNC_LDS 1
#else
#define HAVE_ASYNC_LDS 0
#endif

#if defined(__HIP_DEVICE_COMPILE__) && \
    __has_builtin(__builtin_amdgcn_s_wait_asynccnt)
#define WAIT_ASYNCCNT(n) __builtin_amdgcn_s_wait_asynccnt(n)
#elif defined(__HIP_DEVICE_COMPILE__)
#define WAIT_ASYNCCNT(n) asm volatile("s_wait_asynccnt %0" ::"n"(n) : "memory")
#else
#define WAIT_ASYNCCNT(n)
#endif

// Process 4 consecutive classes of one anchor into the thread-private bins.
__device__ __forceinline__ void ghmc_accum(const v4f x4, const int t,
                                           const int c0,
                                           float2* __restrict__ myrow)
{
    const float vmask = (t > -1) ? 1.0f : 0.0f;       // invalid -> add zeros
    const int selmc0 = ((t >= 1) ? (t - 1) : -1000) - c0;  // onehot when == k
    const float xs[4] = {x4.x, x4.y, x4.z, x4.w};
#pragma unroll
    for (int k = 0; k < 4; ++k) {
        const float x = xs[k];
        const bool oh = (selmc0 == k);
        // Unified form: x' = oh ? -x : x
        //   bce = relu(x') + log1p(exp(-|x'|))   (== reference stable BCE)
        //   g   = sigmoid(x') = |sigmoid(x) - oh|
        const float xp = oh ? -x : x;
        const float u  = __expf(-fabsf(xp));          // shared transcendental
        const float t1 = 1.0f + u;
        const float bce = fmaxf(xp, 0.0f) + __logf(t1);
        const float r  = __builtin_amdgcn_rcpf(t1);   // 1/(1+u)
        const float g  = (xp >= 0.0f) ? r : u * r;    // sigmoid(x')
        int bin = (int)(g * 10.0f);
        bin = bin > (BINS - 1) ? (BINS - 1) : bin;

        // deterministic per-thread LDS RMW (ds_load_b64/ds_store_b64)
        float2 h = myrow[bin];
        h.x += vmask;                                  // count
        h.y += vmask * bce;                            // bce sum
        myrow[bin] = h;
    }
}

__global__ __launch_bounds__(T1) void ghmc_pass1(
    const float* __restrict__ in, const int* __restrict__ tgt,
    float* __restrict__ part, int n4)
{
    __shared__ float2 hist[T1 * PAD];   // 22.5 KB private histograms
#if HAVE_ASYNC_LDS
    __shared__ v4f stage[DEPTH][T1];    // 16 KB async staging ring
    __shared__ v4f dummy[T1];           // 4 KB sink for tail over-issue
#endif
    const int tid = threadIdx.x;
    float2* myrow = &hist[tid * PAD];
#pragma unroll
    for (int b = 0; b < BINS; ++b) myrow[b] = make_float2(0.0f, 0.0f);
    __syncthreads();

    const v4f* in4 = (const v4f*)in;
    const int gid    = blockIdx.x * T1 + tid;
    const int totalT = gridDim.x * T1;
    const int iters  = n4 / totalT;        // wave-uniform main trip count
    const int n4main = iters * totalT;

#if HAVE_ASYNC_LDS
    // gfx1250 path: async global->LDS streaming, 3 loads in flight per wave.
    if (iters > 0) {
        // Prologue: issue loads for iterations 0..2 (clamped to valid work;
        // out-of-range issues are redirected to the dummy slot).
#pragma unroll
        for (int p = 0; p < DEPTH - 1; ++p) {
            const bool real = (p < iters);
            __builtin_amdgcn_global_load_async_to_lds_b128(
                (as1_vi4_ptr)(in4 + (real ? gid + p * totalT : gid)),
                real ? (as3_vi4_ptr)&stage[p][tid]
                     : (as3_vi4_ptr)&dummy[tid],
                0, 0);
        }
#pragma unroll 4
        for (int it = 0; it < iters; ++it) {
            // Always issue exactly one load -> outstanding is at most 4
            // after issue, so a constant s_wait_asynccnt 3 frees buffer `it`.
            const int nx = it + (DEPTH - 1);
            const bool real = (nx < iters);
            __builtin_amdgcn_global_load_async_to_lds_b128(
                (as1_vi4_ptr)(in4 + (real ? gid + nx * totalT : gid)),
                real ? (as3_vi4_ptr)&stage[nx & (DEPTH - 1)][tid]
                     : (as3_vi4_ptr)&dummy[tid],
                0, 0);
            WAIT_ASYNCCNT(DEPTH - 1);        // oldest (current) buffer landed
            asm volatile("" ::: "memory");   // keep LDS consume below the wait
            const v4f x4 = stage[it & (DEPTH - 1)][tid];
            const int i4 = gid + it * totalT;
            const int anchor = i4 / (C_CLS / 4);
            const int c0 = (i4 - anchor * (C_CLS / 4)) * 4;
            ghmc_accum(x4, tgt[anchor], c0, myrow);
        }
        WAIT_ASYNCCNT(0);                    // drain tail dummy issues
        asm volatile("" ::: "memory");
    }
#else
    // Fallback: direct NT b128 stream + gfx1250 prefetch.
    for (int i4 = gid; i4 < n4main; i4 += totalT) {
        __builtin_prefetch((const void*)(in4 + i4 + totalT), 0, 1);
        const v4f x4 = __builtin_nontemporal_load(in4 + i4);
        const int anchor = i4 / (C_CLS / 4);
        const int c0 = (i4 - anchor * (C_CLS / 4)) * 4;
        ghmc_accum(x4, tgt[anchor], c0, myrow);
    }
#endif

    // Scalar tail (empty for the harness shapes: n4 % totalT == 0).
    {
        const int i4 = n4main + gid;
        if (i4 < n4) {
            const v4f x4 = __builtin_nontemporal_load(in4 + i4);
            const int anchor = i4 / (C_CLS / 4);
            const int c0 = (i4 - anchor * (C_CLS / 4)) * 4;
            ghmc_accum(x4, tgt[anchor], c0, myrow);
        }
    }
    __syncthreads();

    // block tree-reduction over the 256 private histograms
#pragma unroll
    for (int off = T1 / 2; off >= 1; off >>= 1) {
        if (tid < off) {
#pragma unroll
            for (int b = 0; b < BINS; ++b) {
                float2 a = hist[tid * PAD + b];
                const float2 c = hist[(tid + off) * PAD + b];
                a.x += c.x; a.y += c.y;
                hist[tid * PAD + b] = a;
            }
        }
        __syncthreads();
    }
    if (tid < BINS) {
        const float2 r0 = hist[tid];                       // row 0, bin=tid
        part[blockIdx.x * (2 * BINS) + tid]        = r0.x; // counts
        part[blockIdx.x * (2 * BINS) + BINS + tid] = r0.y; // sums
    }
}

__global__ __launch_bounds__(320) void ghmc_pass2(
    const float* __restrict__ part, float* __restrict__ out, int nblocks)
{
    const int lane = threadIdx.x & 31;
    const int bin  = threadIdx.x >> 5;   // 10 waves, one bin per wave
    float c = 0.0f, s = 0.0f;
    for (int blk = lane; blk < nblocks; blk += 32) {   // fixed order: deterministic
        c += part[blk * (2 * BINS) + bin];
        s += part[blk * (2 * BINS) + BINS + bin];
    }
#pragma unroll
    for (int off = 16; off >= 1; off >>= 1) {          // wave32 shuffle reduce
        c += __shfl_down(c, off, 32);
        s += __shfl_down(s, off, 32);
    }
    __shared__ float sc[BINS], ss[BINS];
    if (lane == 0) { sc[bin] = c; ss[bin] = s; }
    __syncthreads();
    if (threadIdx.x == 0) {
        int n = 0;
        float acc = 0.0f;
#pragma unroll
        for (int b = 0; b < BINS; ++b) {
            if (sc[b] > 0.0f) { ++n; acc += ss[b] / sc[b]; }
        }
        out[0] = acc / (n > 0 ? (float)n : 1.0f);      // LOSS_WEIGHT = 1
    }
}

extern "C" void kernel_launch(void* const* d_in, const int* in_sizes, int n_in,
                              void* d_out, int out_size, void* d_ws, size_t ws_size,
                              hipStream_t stream) {
    const float* in  = (const float*)d_in[0];   // [B, A, C] f32
    const int*   tgt = (const int*)d_in[1];     // [B, A] i32
    float* out  = (float*)d_out;                // scalar f32
    float* part = (float*)d_ws;                 // per-block partials

    const int n  = in_sizes[0];                 // B*A*C (multiple of 4)
    const int n4 = n / 4;

    int nb = NB_MAX;
    const size_t need = (size_t)nb * 2 * BINS * sizeof(float);
    if (need > ws_size) {
        nb = (int)(ws_size / (2 * BINS * sizeof(float)));
        if (nb < 1) nb = 1;
    }

    ghmc_pass1<<<nb, T1, 0, stream>>>(in, tgt, part, n4);
    ghmc_pass2<<<1, 320, 0, stream>>>(part, out, nb);
}